// MobiusGraphConv_17025250361975
// MI455X (gfx1250) — compile-verified
//
#include <hip/hip_runtime.h>
#include <hip/hip_bf16.h>

typedef __attribute__((ext_vector_type(2))) float v2f;
typedef __attribute__((ext_vector_type(8))) float v8f;

#define N_TOTAL 8192
#define KDIM    16
#define FIN     32
#define FOUT    32

// Kernel-1 decomposition: 16 blocks x 8 waves = 128 partial-T producers.
#define WAVES_PER_BLOCK 8
#define K1_BLOCKS       16
#define K1_WAVES        (K1_BLOCKS * WAVES_PER_BLOCK)   // 128
#define ROWS_PER_WAVE   (N_TOTAL / K1_WAVES)            // 64

// ---------------------------------------------------------------------------
// Kernel 1: partial T = eigenVec^T @ input  per wave, via V_WMMA_F32_16X16X4_F32
// A tile (16x4 f32): A[M=k][c] = ev[nb+c][k]   (M dim = eigen index k)
// B tile (4x16 f32): B[c][f]   = input[nb+c][f]
// ---------------------------------------------------------------------------
__global__ __launch_bounds__(256)
void k1_partialT(const float* __restrict__ ev,
                 const float* __restrict__ in,
                 float* __restrict__ partial) {
    const int wave = blockIdx.x * WAVES_PER_BLOCK + (threadIdx.x >> 5);
    const int lane = threadIdx.x & 31;
    const int m    = lane & 15;      // M index (k) / N index (f) within half
    const int half = lane >> 4;      // 0: K=0,1   1: K=2,3

    v8f acc0 = {};  // f columns 0..15
    v8f acc1 = {};  // f columns 16..31

    const int nstart = wave * ROWS_PER_WAVE;
    #pragma unroll 4
    for (int it = 0; it < ROWS_PER_WAVE / 4; ++it) {
        const int nb = nstart + it * 4 + half * 2;   // this half's first K row
        v2f a, b0, b1;
        a.x  = ev[nb * KDIM + m];
        a.y  = ev[(nb + 1) * KDIM + m];
        b0.x = in[nb * FIN + m];
        b0.y = in[(nb + 1) * FIN + m];
        b1.x = in[nb * FIN + 16 + m];
        b1.y = in[(nb + 1) * FIN + 16 + m];
        acc0 = __builtin_amdgcn_wmma_f32_16x16x4_f32(false, a, false, b0,
                                                     (short)0, acc0, false, false);
        acc1 = __builtin_amdgcn_wmma_f32_16x16x4_f32(false, a, false, b1,
                                                     (short)0, acc1, false, false);
    }

    // C/D layout: VGPR v -> row M = v (lanes 0-15) or v+8 (lanes 16-31), col = m
    float* p = partial + wave * (KDIM * FIN);
    #pragma unroll
    for (int v = 0; v < 8; ++v) {
        const int row = half ? (v + 8) : v;
        p[row * FIN + m]      = acc0[v];
        p[row * FIN + 16 + m] = acc1[v];
    }
}

// ---------------------------------------------------------------------------
// Kernel 2: deterministic reduce of 128 partial T's, Mobius coefficients,
//           S = diag(Mr) @ T @ W0 - diag(Mi) @ T @ W1     [16 x 32]
// ---------------------------------------------------------------------------
__global__ __launch_bounds__(512)
void k2_reduce_S(const float* __restrict__ partial,
                 const float* __restrict__ eigenVal,
                 const float* __restrict__ W,
                 const float* __restrict__ A,
                 const float* __restrict__ B,
                 const float* __restrict__ C,
                 const float* __restrict__ D,
                 float* __restrict__ S) {
    __shared__ float Tl[KDIM * FIN];
    __shared__ float Mr[KDIM];
    __shared__ float Mi[KDIM];
    const int tid = threadIdx.x;   // 0..511 == KDIM*FIN

    float s = 0.0f;
    for (int w = 0; w < K1_WAVES; ++w)
        s += partial[w * (KDIM * FIN) + tid];
    Tl[tid] = s;

    if (tid < KDIM) {
        const float lam = eigenVal[tid];
        const float m1r = A[tid]      * lam + B[tid];
        const float m1i = A[16 + tid] * lam + B[16 + tid];
        const float m2r = 1.0f / (C[tid]      * lam + D[tid]);       // batched real inverse
        const float m2i = 1.0f / (C[16 + tid] * lam + D[16 + tid]);  // of diagonal matrices
        Mr[tid] = m1r * m2r - m1i * m2i;
        Mi[tid] = m1i * m2r + m1r * m2i;
    }
    __syncthreads();

    const int k = tid >> 5;
    const int f = tid & 31;
    float s0 = 0.0f, s1 = 0.0f;
    #pragma unroll
    for (int j = 0; j < FIN; ++j) {
        const float t = Tl[k * FIN + j];
        s0 += t * W[j * FOUT + f];                // T @ W0
        s1 += t * W[FIN * FOUT + j * FOUT + f];   // T @ W1
    }
    S[k * FOUT + f] = Mr[k] * s0 - Mi[k] * s1;
}

// ---------------------------------------------------------------------------
// Kernel 3: out = 2 * eigenVec @ S + bias, one 16-row tile per wave,
//           K=16 folded as 4 chained 16x16x4 f32 WMMAs, 2 tiles for FOUT=32.
// ---------------------------------------------------------------------------
__global__ __launch_bounds__(256)
void k3_out(const float* __restrict__ ev,
            const float* __restrict__ S,
            const float* __restrict__ bias,
            float* __restrict__ out) {
    const int wave = blockIdx.x * WAVES_PER_BLOCK + (threadIdx.x >> 5);
    const int lane = threadIdx.x & 31;
    const int m    = lane & 15;
    const int half = lane >> 4;
    const int n0   = wave * 16;

    v8f acc0 = {};
    v8f acc1 = {};
    #pragma unroll
    for (int kc = 0; kc < 4; ++kc) {
        const int kg = kc * 4 + half * 2;   // this half's first K index
        v2f a, b0, b1;
        a.x  = ev[(n0 + m) * KDIM + kg];
        a.y  = ev[(n0 + m) * KDIM + kg + 1];
        b0.x = S[kg * FOUT + m];
        b0.y = S[(kg + 1) * FOUT + m];
        b1.x = S[kg * FOUT + 16 + m];
        b1.y = S[(kg + 1) * FOUT + 16 + m];
        acc0 = __builtin_amdgcn_wmma_f32_16x16x4_f32(false, a, false, b0,
                                                     (short)0, acc0, false, false);
        acc1 = __builtin_amdgcn_wmma_f32_16x16x4_f32(false, a, false, b1,
                                                     (short)0, acc1, false, false);
    }

    const float bl = bias[m];
    const float bh = bias[16 + m];
    #pragma unroll
    for (int v = 0; v < 8; ++v) {
        const int row = n0 + (half ? (v + 8) : v);
        out[row * FOUT + m]      = 2.0f * acc0[v] + bl;
        out[row * FOUT + 16 + m] = 2.0f * acc1[v] + bh;
    }
}

// ---------------------------------------------------------------------------
extern "C" void kernel_launch(void* const* d_in, const int* in_sizes, int n_in,
                              void* d_out, int out_size, void* d_ws, size_t ws_size,
                              hipStream_t stream) {
    const float* input    = (const float*)d_in[0];   // [8192,32]
    const float* eigenVal = (const float*)d_in[1];   // [16]
    const float* eigenVec = (const float*)d_in[2];   // [8192,16]
    const float* W        = (const float*)d_in[3];   // [2,32,32]
    const float* A        = (const float*)d_in[4];   // [2,16]
    const float* B        = (const float*)d_in[5];
    const float* C        = (const float*)d_in[6];
    const float* D        = (const float*)d_in[7];
    const float* bias     = (const float*)d_in[8];   // [32]
    float* out = (float*)d_out;

    float* partial = (float*)d_ws;                        // 128 * 512 floats = 256 KB
    float* S       = partial + (size_t)K1_WAVES * KDIM * FIN;  // 512 floats

    k1_partialT<<<K1_BLOCKS, 256, 0, stream>>>(eigenVec, input, partial);
    k2_reduce_S<<<1, 512, 0, stream>>>(partial, eigenVal, W, A, B, C, D, S);
    k3_out<<<N_TOTAL / (16 * WAVES_PER_BLOCK), 256, 0, stream>>>(eigenVec, S, bias, out);
}